// Net1_3899830305164
// MI455X (gfx1250) — compile-verified
//
#include <hip/hip_runtime.h>
#include <math.h>

// ---------------------------------------------------------------------------
// GCN (2x GCNConv + Linear + log_softmax) for gfx1250 / MI455X.
// Bandwidth-bound (whole problem lives in 192MB L2); GEMMs done with
// V_WMMA_F32_16X16X4_F32 (fp32 WMMA, wave32) to keep fp32 precision.
// ---------------------------------------------------------------------------

typedef __attribute__((ext_vector_type(2))) float v2f;
typedef __attribute__((ext_vector_type(8))) float v8f;

#define FEAT 64   // NUM_FEATURES == HIDDEN == 64
#define NCLS 16

// ---------------- degree / normalization ----------------------------------

__global__ __launch_bounds__(256) void deg_init_kernel(int* __restrict__ deg, int n) {
  int i = blockIdx.x * blockDim.x + threadIdx.x;
  if (i < n) deg[i] = 1;  // self-loop
}

__global__ __launch_bounds__(256) void deg_edges_kernel(int* __restrict__ deg,
                                                        const int* __restrict__ dst, int e) {
  int i = blockIdx.x * blockDim.x + threadIdx.x;
  if (i < e) atomicAdd(&deg[dst[i]], 1);
}

__global__ __launch_bounds__(256) void deg_rsqrt_kernel(float* __restrict__ dis, int n) {
  int i = blockIdx.x * blockDim.x + threadIdx.x;
  if (i < n) {
    int d = ((const int*)dis)[i];
    dis[i] = rsqrtf((float)d);
  }
}

// ---------------- dense GEMM: C[M,64] = A[M,64] @ W[64,64] (fp32 WMMA) -----
// One wave per 16x16 output tile; K stepped by 4 -> 16 v_wmma_f32_16x16x4_f32.
// A frag (16x4 f32, 2 VGPR): lane&15 = M row, lane>>4 selects K-pair {0,1|2,3}.
// C/D (8 VGPR): lanes 0-15 rows 0-7, lanes 16-31 rows 8-15, col = lane&15.
// B frag assumed mirror of A: lane&15 = N col, lane>>4 selects K-pair.

__global__ __launch_bounds__(256) void gemm64_wmma_kernel(const float* __restrict__ A,
                                                          const float* __restrict__ W,
                                                          float* __restrict__ C, int M) {
  int wave = (int)((blockIdx.x * blockDim.x + threadIdx.x) >> 5);
  int lane = threadIdx.x & 31;
  int nTile = wave & 3;   // 64/16 = 4 tiles along N
  int mTile = wave >> 2;
  if (mTile * 16 >= M) return;           // wave-uniform; EXEC stays all-ones
  const int m0 = mTile * 16, n0 = nTile * 16;
  const int lm = lane & 15;
  const int kb = (lane >> 4) << 1;       // 0 or 2: K offset within step-of-4

  v8f c = {};
  const float* arow = A + (size_t)(m0 + lm) * FEAT + kb;    // 8B-aligned
  const float* wcol = W + (size_t)kb * FEAT + n0 + lm;
#pragma unroll
  for (int k0 = 0; k0 < FEAT; k0 += 4) {
    v2f a, b;
    const float2 av = *(const float2*)(arow + k0);          // A[m][k0+kb .. +1]
    a.x = av.x; a.y = av.y;
    b.x = wcol[(size_t)k0 * FEAT];                          // W[k0+kb  ][n]
    b.y = wcol[(size_t)k0 * FEAT + FEAT];                   // W[k0+kb+1][n]
    c = __builtin_amdgcn_wmma_f32_16x16x4_f32(false, a, false, b,
                                              (short)0, c, false, false);
  }
  const int rbase = m0 + (lane >> 4) * 8;
  const int col = n0 + lm;
#pragma unroll
  for (int v = 0; v < 8; ++v)
    C[(size_t)(rbase + v) * FEAT + col] = c[v];
}

// ---------------- aggregation ----------------------------------------------
// agg[i][f] = b[f] + dis[i]^2 * h[i][f]   (bias + self-loop message)

__global__ __launch_bounds__(256) void agg_init_kernel(float* __restrict__ agg,
                                                       const float* __restrict__ h,
                                                       const float* __restrict__ dis,
                                                       const float* __restrict__ bias,
                                                       int n) {
  int t = blockIdx.x * blockDim.x + threadIdx.x;
  if (t < n * FEAT) {
    int node = t >> 6, f = t & 63;
    float d = dis[node];
    agg[t] = bias[f] + d * d * h[t];
  }
}

// One wave32 per edge; each lane handles a float2 of the 64-wide feature row.
// Gather of h[src] is a coalesced 256B row read; scatter via global fp32 atomics.
__global__ __launch_bounds__(256) void agg_edges_kernel(float* __restrict__ agg,
                                                        const float* __restrict__ h,
                                                        const float* __restrict__ dis,
                                                        const int* __restrict__ src,
                                                        const int* __restrict__ dst, int e) {
  int wave = (int)((blockIdx.x * blockDim.x + threadIdx.x) >> 5);
  int lane = threadIdx.x & 31;
  if (wave >= e) return;
  int s = src[wave], d = dst[wave];
  float w = dis[s] * dis[d];
  float2 v = ((const float2*)(h + (size_t)s * FEAT))[lane];
  float* ad = agg + (size_t)d * FEAT + lane * 2;
  unsafeAtomicAdd(ad + 0, w * v.x);   // -> global_atomic_add_f32
  unsafeAtomicAdd(ad + 1, w * v.y);
}

__global__ __launch_bounds__(256) void relu_kernel(float* __restrict__ out,
                                                   const float* __restrict__ in, int n) {
  int t = blockIdx.x * blockDim.x + threadIdx.x;
  if (t < n) out[t] = fmaxf(in[t], 0.0f);
}

// ---------------- final linear + log_softmax -------------------------------
// z = h @ W3 + b3 (64x16, W3 fits in cache), then row log_softmax over 16.

__global__ __launch_bounds__(256) void fc_logsoftmax_kernel(const float* __restrict__ h,
                                                            const float* __restrict__ W3,
                                                            const float* __restrict__ b3,
                                                            float* __restrict__ out, int n) {
  int node = blockIdx.x * blockDim.x + threadIdx.x;
  if (node >= n) return;
  const float* hr = h + (size_t)node * FEAT;
  float z[NCLS];
#pragma unroll
  for (int c = 0; c < NCLS; ++c) z[c] = b3[c];
#pragma unroll 4
  for (int k = 0; k < FEAT; ++k) {
    float hv = hr[k];
    const float* wr = W3 + k * NCLS;
#pragma unroll
    for (int c = 0; c < NCLS; ++c) z[c] = fmaf(hv, wr[c], z[c]);
  }
  float m = z[0];
#pragma unroll
  for (int c = 1; c < NCLS; ++c) m = fmaxf(m, z[c]);
  float s = 0.0f;
#pragma unroll
  for (int c = 0; c < NCLS; ++c) s += expf(z[c] - m);
  float lse = m + logf(s);
  float* o = out + (size_t)node * NCLS;
#pragma unroll
  for (int c = 0; c < NCLS; ++c) o[c] = z[c] - lse;
}

// ---------------------------------------------------------------------------

extern "C" void kernel_launch(void* const* d_in, const int* in_sizes, int n_in,
                              void* d_out, int out_size, void* d_ws, size_t ws_size,
                              hipStream_t stream) {
  const float* x  = (const float*)d_in[0];
  const int*   ei = (const int*)d_in[1];
  const float* W1 = (const float*)d_in[2];
  const float* b1 = (const float*)d_in[3];
  const float* W2 = (const float*)d_in[4];
  const float* b2 = (const float*)d_in[5];
  const float* W3 = (const float*)d_in[6];
  const float* b3 = (const float*)d_in[7];

  const int N = in_sizes[0] / FEAT;     // 100000
  const int E = in_sizes[1] / 2;        // 1600000
  const int* src = ei;                  // edge_index[0]
  const int* dst = ei + E;              // edge_index[1]

  // workspace: dis[N] | buf0[N*64] | buf1[N*64]
  char* ws = (char*)d_ws;
  size_t offDis = 0;
  size_t offB0  = ((size_t)N * 4 + 255) & ~(size_t)255;
  size_t offB1  = offB0 + (size_t)N * FEAT * 4;
  float* dis  = (float*)(ws + offDis);
  float* buf0 = (float*)(ws + offB0);
  float* buf1 = (float*)(ws + offB1);

  float* out_logp = (float*)d_out;                    // [N,16]
  float* out_h    = (float*)d_out + (size_t)N * NCLS; // [N,64]

  const int T = 256;
  int bN   = (N + T - 1) / T;
  int bE   = (E + T - 1) / T;
  int bNF  = (N * FEAT + T - 1) / T;
  int bGem = (((size_t)(N / 16) * 4 * 32) + T - 1) / T;  // N%16==0 for N=100000... (6250*4 waves)
  int bEdg = (E + 7) / 8;                                // 8 waves/block, 1 wave/edge

  // ---- normalization: deg -> d^{-1/2} ----
  deg_init_kernel<<<bN, T, 0, stream>>>((int*)dis, N);
  deg_edges_kernel<<<bE, T, 0, stream>>>((int*)dis, dst, E);
  deg_rsqrt_kernel<<<bN, T, 0, stream>>>(dis, N);

  // ---- layer 1: h1 = relu(Agg(x @ W1) + b1) ----
  gemm64_wmma_kernel<<<bGem, T, 0, stream>>>(x, W1, buf0, N);
  agg_init_kernel<<<bNF, T, 0, stream>>>(buf1, buf0, dis, b1, N);
  agg_edges_kernel<<<bEdg, T, 0, stream>>>(buf1, buf0, dis, src, dst, E);
  relu_kernel<<<bNF, T, 0, stream>>>(buf1, buf1, N * FEAT);

  // ---- layer 2: h2 = relu(Agg(h1 @ W2) + b2) ----
  gemm64_wmma_kernel<<<bGem, T, 0, stream>>>(buf1, W2, buf0, N);
  agg_init_kernel<<<bNF, T, 0, stream>>>(buf1, buf0, dis, b2, N);
  agg_edges_kernel<<<bEdg, T, 0, stream>>>(buf1, buf0, dis, src, dst, E);
  relu_kernel<<<bNF, T, 0, stream>>>(out_h, buf1, N * FEAT);   // second tuple output

  // ---- head: log_softmax(h2 @ W3 + b3) ----
  fc_logsoftmax_kernel<<<bN, T, 0, stream>>>(out_h, W3, b3, out_logp, N);
}